// DistMultDecoder_25074019074708
// MI455X (gfx1250) — compile-verified
//
#include <hip/hip_runtime.h>

typedef __attribute__((ext_vector_type(2))) float v2f;
typedef __attribute__((ext_vector_type(4))) float v4f;
typedef __attribute__((ext_vector_type(8))) float v8f;

#define HIDDEN 128

// One wave32 handles 16 edges. Scores for the tile are the diagonal of a
// 16x16 f32 WMMA accumulation: D += A(16x4) x B(4x16), A = (z_src .* z_dst)
// column chunks, B = rel_emb column chunks. Column->K-slot assignment is a
// permutation applied identically to A and B, so each lane can load one
// contiguous float4 per row per iteration (lanes 0-15: cols 8t..8t+3,
// lanes 16-31: cols 8t+4..8t+7) and issue two WMMAs.
__global__ __launch_bounds__(256) void distmult_wmma_kernel(
    const float* __restrict__ z,
    const float* __restrict__ rel,
    const int*   __restrict__ edge_index,   // [2, E] flat
    const int*   __restrict__ edge_type,    // [E]
    float*       __restrict__ out,          // [E]
    int nEdges, int numTiles)
{
    const int lane = threadIdx.x & 31;
    const int waveInBlock = threadIdx.x >> 5;
    const int tile = blockIdx.x * (blockDim.x >> 5) + waveInBlock;
    if (tile >= numTiles) return;          // wave-uniform: EXEC stays all-1s

    const int m = lane & 15;               // edge-in-tile (row for A, col for B)
    const int e = tile * 16 + m;
    const int src = edge_index[e];
    const int dst = edge_index[nEdges + e];
    const int typ = edge_type[e];

    const float* zs = z   + (size_t)src * HIDDEN;
    const float* zd = z   + (size_t)dst * HIDDEN;
    const float* rr = rel + (size_t)typ * HIDDEN;

    const int colBase = (lane >> 4) << 2;  // 0 for lanes 0-15, 4 for lanes 16-31

    v8f acc = {};

#pragma unroll
    for (int t = 0; t < HIDDEN / 8; ++t) {
        const int c0 = t * 8 + colBase;
        v4f s = *(const v4f*)(zs + c0);
        v4f d = *(const v4f*)(zd + c0);
        v4f r = *(const v4f*)(rr + c0);
        v4f w = s * d;                     // elementwise z_src .* z_dst

        v2f a0 = { w.x, w.y };
        v2f b0 = { r.x, r.y };
        acc = __builtin_amdgcn_wmma_f32_16x16x4_f32(
            false, a0, false, b0, (short)0, acc, false, false);

        v2f a1 = { w.z, w.w };
        v2f b1 = { r.z, r.w };
        acc = __builtin_amdgcn_wmma_f32_16x16x4_f32(
            false, a1, false, b1, (short)0, acc, false, false);
    }

    // Extract diagonal of D (16x16 f32 C/D layout):
    //   D[v][v]     -> VGPR v,   lane v       (v = 0..7)
    //   D[v+8][v+8] -> VGPR v,   lane v+24
    float score = 0.0f;
#pragma unroll
    for (int v = 0; v < 8; ++v) {
        float lo = __shfl(acc[v], v,      32);
        float hi = __shfl(acc[v], v + 24, 32);
        if (lane == v)     score = lo;
        if (lane == v + 8) score = hi;
    }
    if (lane < 16) out[tile * 16 + lane] = score;
}

// Scalar tail for E % 16 edges (E = 600000 -> empty, kept for safety).
__global__ void distmult_tail_kernel(
    const float* __restrict__ z,
    const float* __restrict__ rel,
    const int*   __restrict__ edge_index,
    const int*   __restrict__ edge_type,
    float*       __restrict__ out,
    int nEdges, int start)
{
    int e = start + blockIdx.x * blockDim.x + threadIdx.x;
    if (e >= nEdges) return;
    const float* zs = z   + (size_t)edge_index[e] * HIDDEN;
    const float* zd = z   + (size_t)edge_index[nEdges + e] * HIDDEN;
    const float* rr = rel + (size_t)edge_type[e] * HIDDEN;
    float s = 0.0f;
    for (int d = 0; d < HIDDEN; ++d) s += zs[d] * zd[d] * rr[d];
    out[e] = s;
}

extern "C" void kernel_launch(void* const* d_in, const int* in_sizes, int n_in,
                              void* d_out, int out_size, void* d_ws, size_t ws_size,
                              hipStream_t stream) {
    const float* z          = (const float*)d_in[0];
    const float* rel        = (const float*)d_in[1];
    const int*   edge_index = (const int*)d_in[2];
    const int*   edge_type  = (const int*)d_in[3];
    float*       out        = (float*)d_out;

    const int nEdges   = in_sizes[3];     // edge_type has E elements
    const int numTiles = nEdges >> 4;
    const int rem      = nEdges & 15;

    if (numTiles > 0) {
        const int wavesPerBlock = 8;      // 256 threads = 8 wave32
        const int blocks = (numTiles + wavesPerBlock - 1) / wavesPerBlock;
        distmult_wmma_kernel<<<blocks, 256, 0, stream>>>(
            z, rel, edge_index, edge_type, out, nEdges, numTiles);
    }
    if (rem > 0) {
        distmult_tail_kernel<<<1, 32, 0, stream>>>(
            z, rel, edge_index, edge_type, out, nEdges, numTiles * 16);
    }
}